// PCEN_24215025615500
// MI455X (gfx1250) — compile-verified
//
#include <hip/hip_runtime.h>
#include <math.h>

// ---- problem constants (from reference) ----
#define B_   16
#define C_   2
#define F_   257            // FREQ/2
#define T_   4000
#define ROWS (B_*C_*F_)     // 8224 mag rows
#define TILES (ROWS/16)     // 514 (exact)
#define WAVES_PER_BLOCK 8
#define LDS_STRIDE 20       // 16 + 4 pad floats; 20n mod 64 distinct for n=0..15
#define FLOORV 1e-6f

typedef float v8f __attribute__((ext_vector_type(8)));
typedef float v2f __attribute__((ext_vector_type(2)));

// fast x^y for x > 0: v_exp_f32(y * v_log_f32(x))
__device__ __forceinline__ float fpow(float x, float y) {
  return __builtin_amdgcn_exp2f(y * __builtin_amdgcn_logf(x));
}

// out = mag * (FLOOR+ema)^(-a) + d, then ^(1/root), minus d^(1/root)
__device__ __forceinline__ float pcen_ep(float mag, float ema,
                                         float naexp, float dd, float oor, float dpow) {
  float deninv = fpow(FLOORV + ema, naexp);             // (FLOOR+ema)^(-a)
  return fpow(fmaf(mag, deninv, dd), oor) - dpow;
}

__global__ __launch_bounds__(256) void pcen_wmma_kernel(
    const float* __restrict__ tf, const float* __restrict__ weights,
    const float* __restrict__ alpha, const float* __restrict__ delta,
    const float* __restrict__ root, float* __restrict__ out)
{
  __shared__ float smem_all[WAVES_PER_BLOCK][16 * LDS_STRIDE];

  const int lane = threadIdx.x & 31;
  const int wv   = threadIdx.x >> 5;
  const int tile = blockIdx.x * WAVES_PER_BLOCK + wv;
  if (tile >= TILES) return;
  float* smem = smem_all[wv];

  const int rowBase = tile * 16;
  const int half = lane >> 4;      // 0: lanes 0-15, 1: lanes 16-31
  const int n    = lane & 15;      // matrix column = row-within-tile

  // ---- loader mapping: 2 lanes per row, 8 consecutive t each ----
  const int lrow = lane >> 1;
  const int ltof = (lane & 1) * 8;
  const int grow = rowBase + lrow;
  const int gb   = grow / (C_ * F_);
  const int grem = grow % (C_ * F_);
  const float* reP = tf + ((size_t)(gb * C_ + grem / F_) * (2 * F_) + (grem % F_)) * T_;
  const float* imP = reP + (size_t)F_ * T_;

  // ---- epilogue per-lane channel params (row = rowBase + n) ----
  const int erow = rowBase + n;
  const int ec   = (erow / F_) % C_;
  const float naexp = -fminf(alpha[ec], 1.0f);          // negated exponent
  const float oor   = 1.0f / fmaxf(root[ec], 1.0f);
  const float dd    = delta[ec];
  const float dpow  = fpow(dd, oor);
  float* outP = out + (size_t)erow * T_;

  const float w = fminf(fmaxf(weights[0], 0.0f), 1.0f);
  const float q = 1.0f - w;

  // ---- A-matrix (L, lower triangular) per ISA 16x4 f32 A layout ----
  // VGPR0 holds K=4b+0 (lanes 0-15) / K=4b+2 (lanes 16-31); VGPR1: +1
  float A0[4], A1[4];
#pragma unroll
  for (int b = 0; b < 4; ++b) {
    const int k0 = 4 * b + 2 * half;
    A0[b] = (k0     <= n) ? w * fpow(q, (float)(n - k0))     : 0.0f;
    A1[b] = (k0 + 1 <= n) ? w * fpow(q, (float)(n - k0 - 1)) : 0.0f;
  }
  // carry coefficients q^{m+1}, m = i + 8*half (C/D layout time index)
  float qp[8];
  qp[0] = fpow(q, (float)(1 + 8 * half));
#pragma unroll
  for (int i = 1; i < 8; ++i) qp[i] = qp[i - 1] * q;

  float carry = 0.0f;

  for (int chunk = 0; chunk < T_ / 16; ++chunk) {
    const int t0 = chunk * 16;

    // ---- stage mag tile (16 rows x 16 t) into LDS ----
    const float4 r0 = *(const float4*)(reP + t0 + ltof);
    const float4 r1 = *(const float4*)(reP + t0 + ltof + 4);
    const float4 i0 = *(const float4*)(imP + t0 + ltof);
    const float4 i1 = *(const float4*)(imP + t0 + ltof + 4);
    __builtin_prefetch(reP + t0 + 256, 0, 0);   // global_prefetch_b8, stream ahead
    __builtin_prefetch(imP + t0 + 256, 0, 0);
    float4 m0, m1;
    m0.x = __builtin_amdgcn_sqrtf(fmaf(r0.x, r0.x, i0.x * i0.x));
    m0.y = __builtin_amdgcn_sqrtf(fmaf(r0.y, r0.y, i0.y * i0.y));
    m0.z = __builtin_amdgcn_sqrtf(fmaf(r0.z, r0.z, i0.z * i0.z));
    m0.w = __builtin_amdgcn_sqrtf(fmaf(r0.w, r0.w, i0.w * i0.w));
    m1.x = __builtin_amdgcn_sqrtf(fmaf(r1.x, r1.x, i1.x * i1.x));
    m1.y = __builtin_amdgcn_sqrtf(fmaf(r1.y, r1.y, i1.y * i1.y));
    m1.z = __builtin_amdgcn_sqrtf(fmaf(r1.z, r1.z, i1.z * i1.z));
    m1.w = __builtin_amdgcn_sqrtf(fmaf(r1.w, r1.w, i1.w * i1.w));
    *(float4*)(smem + lrow * LDS_STRIDE + ltof)     = m0;
    *(float4*)(smem + lrow * LDS_STRIDE + ltof + 4) = m1;
    asm volatile("s_wait_dscnt 0" ::: "memory");  // LDS RAW fence (per-wave tile)

    if (chunk == 0) carry = smem[n * LDS_STRIDE];  // init = x0 per reference

    // ---- carry term: acc[i] = q^{m+1} * carry ----
    v8f acc;
#pragma unroll
    for (int i = 0; i < 8; ++i) acc[i] = qp[i] * carry;

    // ---- E = L @ X + carry via 4 chained f32 WMMAs (K=4 each) ----
#pragma unroll
    for (int b = 0; b < 4; ++b) {
      const int k0 = 4 * b + 2 * half;
      v2f Bv, Av;
      Bv.x = smem[n * LDS_STRIDE + k0];       // X[K][n] = mag[row n][t0+K]
      Bv.y = smem[n * LDS_STRIDE + k0 + 1];
      Av.x = A0[b];
      Av.y = A1[b];
      acc = __builtin_amdgcn_wmma_f32_16x16x4_f32(
          false, Av, false, Bv, (short)0, acc, false, false);
    }

    // ---- next carry: ema[t0+15][n] lives in lane 16+n, VGPR 7 ----
    carry = __shfl(acc[7], 16 + n, 32);

    // ---- epilogue: this lane owns (row n, t = t0 + 8*half + i) ----
    const float4 g0 = *(const float4*)(smem + n * LDS_STRIDE + 8 * half);
    const float4 g1 = *(const float4*)(smem + n * LDS_STRIDE + 8 * half + 4);
    float4 o0, o1;
    o0.x = pcen_ep(g0.x, acc[0], naexp, dd, oor, dpow);
    o0.y = pcen_ep(g0.y, acc[1], naexp, dd, oor, dpow);
    o0.z = pcen_ep(g0.z, acc[2], naexp, dd, oor, dpow);
    o0.w = pcen_ep(g0.w, acc[3], naexp, dd, oor, dpow);
    o1.x = pcen_ep(g1.x, acc[4], naexp, dd, oor, dpow);
    o1.y = pcen_ep(g1.y, acc[5], naexp, dd, oor, dpow);
    o1.z = pcen_ep(g1.z, acc[6], naexp, dd, oor, dpow);
    o1.w = pcen_ep(g1.w, acc[7], naexp, dd, oor, dpow);
    *(float4*)(outP + t0 + 8 * half)     = o0;
    *(float4*)(outP + t0 + 8 * half + 4) = o1;
  }
}

extern "C" void kernel_launch(void* const* d_in, const int* in_sizes, int n_in,
                              void* d_out, int out_size, void* d_ws, size_t ws_size,
                              hipStream_t stream) {
  (void)in_sizes; (void)n_in; (void)d_ws; (void)ws_size; (void)out_size;
  const float* tf      = (const float*)d_in[0];
  const float* weights = (const float*)d_in[1];
  const float* alpha   = (const float*)d_in[2];
  const float* delta   = (const float*)d_in[3];
  const float* root    = (const float*)d_in[4];
  float* out = (float*)d_out;

  const int blocks = (TILES + WAVES_PER_BLOCK - 1) / WAVES_PER_BLOCK;  // 65
  pcen_wmma_kernel<<<blocks, 32 * WAVES_PER_BLOCK, 0, stream>>>(
      tf, weights, alpha, delta, root, out);
}